// StateSpaceLayer_48507360641377
// MI455X (gfx1250) — compile-verified
//
#include <hip/hip_runtime.h>

// ---------------------------------------------------------------------------
// StateSpaceLayer for MI455X (gfx1250).
// Chunked-scan reformulation of the FFT causal conv (exact: fft_size=2L means
// no circular wraparound), fp32 WMMA (V_WMMA_F32_16X16X4_F32) for the shared
// GEMM pieces, VALU Toeplitz for the per-channel intra-chunk part.
// u-chunk streaming uses double-buffered ASYNC global->LDS loads (ASYNCcnt)
// when the toolchain exposes the gfx1250 builtins.
// ---------------------------------------------------------------------------

typedef float v2f __attribute__((ext_vector_type(2)));
typedef float v8f __attribute__((ext_vector_type(8)));
typedef int   v4i __attribute__((ext_vector_type(4)));

typedef __attribute__((address_space(1))) v4i* gv4i_p;  // global (device) int4*
typedef __attribute__((address_space(3))) v4i* lv4i_p;  // LDS int4*

#define BATCH   4
#define SEQ_L   4096
#define D_MODEL 1024
#define N_STATE 64
#define CHUNK   64
#define NCHUNK  (SEQ_L / CHUNK)

// workspace layout (floats)
#define WS_R    0        // r[n] = exp(delta0*A)          (64)
#define WS_RT   64       // r^64                           (64)
#define WS_COEF 128      // (A_bar_s-1)/A_safe             (64)
#define WS_DR   192      // delta0*A                       (64)
#define WS_W    256      // W[tau][n] = r^{tau+1}          (64*64)
#define WS_V    4352     // V[j][n]   = r^{63-j}           (64*64)
#define WS_K    8448     // Ksub[k][d] = sum_n r^k C[d,n]  (64*1024)

#if __has_builtin(__builtin_amdgcn_global_load_async_to_lds_b128) && \
    __has_builtin(__builtin_amdgcn_s_wait_asynccnt)
#define USE_ASYNC_LDS 1
#endif

static __device__ __forceinline__ v8f wmma_f32(v2f a, v2f b, v8f c) {
  return __builtin_amdgcn_wmma_f32_16x16x4_f32(false, a, false, b, (short)0, c,
                                               false, false);
}

// Issue loads of one 64x64 u-chunk slice into an LDS buffer.
// Each of the 256 threads covers 4 float4's (64B): whole tile = 16KB.
static __device__ __forceinline__ void chunk_load(const float* __restrict__ gsrc,
                                                  float* ldst, int tid) {
#pragma unroll
  for (int r = 0; r < 4; ++r) {
    int flat = tid + r * 256;          // 0..1023 float4's
    int j = flat >> 4;                 // time row within chunk
    int q = (flat & 15) * 4;           // channel within slice
#if defined(USE_ASYNC_LDS)
    __builtin_amdgcn_global_load_async_to_lds_b128(
        (gv4i_p)(const void*)(gsrc + (size_t)j * D_MODEL + q),
        (lv4i_p)(void*)(ldst + j * 64 + q),
        0, 0);
#else
    float4 uv = *(const float4*)(gsrc + (size_t)j * D_MODEL + q);
    *(float4*)(ldst + j * 64 + q) = uv;
#endif
  }
}

// ---------------------------------------------------------------------------
// Kernel 1: scalar tables (r, r^64, coef, dr, W, V)
// ---------------------------------------------------------------------------
__global__ __launch_bounds__(256) void ssm_setup(const float* __restrict__ A_log,
                                                 const float* __restrict__ delta,
                                                 float* __restrict__ ws) {
  __shared__ float red[256];
  __shared__ float sdr[64];
  int tid = threadIdx.x;
  float s = delta[tid] + delta[tid + 256] + delta[tid + 512] + delta[tid + 768];
  red[tid] = s;
  __syncthreads();
  for (int off = 128; off > 0; off >>= 1) {
    if (tid < off) red[tid] += red[tid + off];
    __syncthreads();
  }
  float dmean = red[0] * (1.0f / 1024.0f);
  float d0 = delta[0];
  if (tid < 64) {
    float A = -expf(A_log[tid]);
    float dr = d0 * A;
    sdr[tid] = dr;
    ws[WS_R + tid]  = expf(dr);
    ws[WS_RT + tid] = expf(64.0f * dr);
    float A_safe = A - 1e-8f;            // A < 0 always -> sign(A) = -1
    float abar = expf(dmean * A);
    ws[WS_COEF + tid] = (abar - 1.0f) / A_safe;
    ws[WS_DR + tid] = dr;
  }
  __syncthreads();
  for (int k = 0; k < 16; ++k) {
    int idx = tid + k * 256;             // 0..4095
    int row = idx >> 6;
    int n = idx & 63;
    float dr = sdr[n];
    ws[WS_W + idx] = expf((float)(row + 1) * dr);   // r^{tau+1}
    ws[WS_V + idx] = expf((float)(63 - row) * dr);  // r^{63-j}
  }
}

// ---------------------------------------------------------------------------
// Kernel 2: Ksub[k,d] = sum_n exp(k*dr_n) * C[d,n]   via WMMA f32 16x16x4
//   A = P[64 x 64] (M=k, K=n), B = C^T (K=n, N=d), out 64 x 1024.
//   256 tiles of 16x16; 8 waves/block, 1 tile/wave, 32 blocks.
// ---------------------------------------------------------------------------
__global__ __launch_bounds__(256) void ssm_ksub(const float* __restrict__ Cmat,
                                                float* __restrict__ ws) {
  __shared__ float sdr[64];
  int tid = threadIdx.x;
  if (tid < 64) sdr[tid] = ws[WS_DR + tid];
  __syncthreads();

  int w = tid >> 5, lane = tid & 31, hf = lane >> 4, m = lane & 15;
  int t = blockIdx.x * 8 + w;
  int ti = t & 3;        // k-tile (0..3)
  int di = t >> 2;       // d-tile (0..63)
  float mrow = (float)(ti * 16 + m);
  int dcol = di * 16 + m;

  v8f acc = {};
#pragma unroll
  for (int kk = 0; kk < 16; ++kk) {
    int k0 = kk * 4 + hf * 2;
    v2f a, b;
    a.x = expf(mrow * sdr[k0]);
    a.y = expf(mrow * sdr[k0 + 1]);
    b.x = Cmat[(size_t)dcol * 64 + k0];
    b.y = Cmat[(size_t)dcol * 64 + k0 + 1];
    acc = wmma_f32(a, b, acc);
  }
#pragma unroll
  for (int i = 0; i < 8; ++i) {
    int krow = ti * 16 + i + hf * 8;
    ws[WS_K + (size_t)krow * 1024 + dcol] = acc[i];
  }
}

// ---------------------------------------------------------------------------
// Kernel 3: main chunked scan.
//   Block = (b, 64-channel slice); 8 waves; h state carried in WMMA C regs.
//   Per chunk: y = W @ (C.*h)^T  (WMMA)  +  intra-chunk Toeplitz (VALU/LDS),
//              h = rT.*h + u^T @ V (WMMA, fused into accumulator).
//   u chunks double-buffered via async global->LDS; next chunk streams in
//   while the current chunk computes.
//   Writes pre-LayerNorm y (with D-skip, freq_align scale, reward@t=0).
// ---------------------------------------------------------------------------
__global__ __launch_bounds__(256) void ssm_main(
    const float* __restrict__ u, const float* __restrict__ Cmat,
    const float* __restrict__ Dvec, const float* __restrict__ freq_align,
    const float* __restrict__ reward, const float* __restrict__ ws,
    float* __restrict__ y_out) {
  __shared__ float sK[64 * 64];      // Ksub slice [k][dd]
  __shared__ float sUbuf[2][64 * 64];// u chunk    [j][dd], double buffered
  __shared__ float sG[64 * 64];      // g = C.*h   [dd][n]

  int tid = threadIdx.x;
  int b = blockIdx.x >> 4;
  int d0 = (blockIdx.x & 15) * 64;
  int w = tid >> 5, lane = tid & 31, hf = lane >> 4, m = lane & 15;

  // preload Ksub slice
  for (int k = 0; k < 16; ++k) {
    int idx = tid + k * 256;
    int kr = idx >> 6, dd = idx & 63;
    sK[idx] = ws[WS_K + (size_t)kr * 1024 + d0 + dd];
  }

  const float* ub = u + (size_t)b * SEQ_L * D_MODEL + d0;
  // prologue: chunk 0 into buffer 0
  chunk_load(ub, sUbuf[0], tid);

  // two 16x16 tiles per wave (ids w and w+8); rows = t>>2, cols = t&3
  int ta = w, tb = w + 8;
  int ri_a = ta >> 2, ci_a = ta & 3;
  int ri_b = tb >> 2, ci_b = tb & 3;

  // chunk-invariant fragments in registers:
  //   W A-frags (GEMM1 rows = tau tile ri), V B-frags (GEMM2 cols = n tile ci)
  float Wfa[16][2], Wfb[16][2], Vfa[16][2], Vfb[16][2];
#pragma unroll
  for (int kk = 0; kk < 16; ++kk) {
    int k0 = kk * 4 + hf * 2;
    Wfa[kk][0] = ws[WS_W + (ri_a * 16 + m) * 64 + k0];
    Wfa[kk][1] = ws[WS_W + (ri_a * 16 + m) * 64 + k0 + 1];
    Wfb[kk][0] = ws[WS_W + (ri_b * 16 + m) * 64 + k0];
    Wfb[kk][1] = ws[WS_W + (ri_b * 16 + m) * 64 + k0 + 1];
    Vfa[kk][0] = ws[WS_V + (k0)     * 64 + ci_a * 16 + m];
    Vfa[kk][1] = ws[WS_V + (k0 + 1) * 64 + ci_a * 16 + m];
    Vfb[kk][0] = ws[WS_V + (k0)     * 64 + ci_b * 16 + m];
    Vfb[kk][1] = ws[WS_V + (k0 + 1) * 64 + ci_b * 16 + m];
  }
  // per-tile channel constants (d = d0 + ci*16 + m)
  float FAa = freq_align[d0 + ci_a * 16 + m];
  float FAb = freq_align[d0 + ci_b * 16 + m];
  float RWa = reward[d0 + ci_a * 16 + m];
  float RWb = reward[d0 + ci_b * 16 + m];
  float Dva = Dvec[d0 + ci_a * 16 + m];
  float Dvb = Dvec[d0 + ci_b * 16 + m];
  // per-tile n constant for state decay (n = ci*16 + m)
  float rta = ws[WS_RT + ci_a * 16 + m];
  float rtb = ws[WS_RT + ci_b * 16 + m];

  v8f hA = {}, hB = {};   // state h[d,n] tiles, carried across chunks

  for (int c = 0; c < NCHUNK; ++c) {
    float* sU = sUbuf[c & 1];
    // ---- stream next chunk into the other buffer (async, overlaps compute) ----
    if (c + 1 < NCHUNK)
      chunk_load(ub + (size_t)(c + 1) * CHUNK * D_MODEL, sUbuf[(c & 1) ^ 1], tid);

    // ---- g[d,n] = C[d,n] * h[d,n]  (from register state) ----
#pragma unroll
    for (int i = 0; i < 8; ++i) {
      int dra = ri_a * 16 + i + hf * 8, nca = ci_a * 16 + m;
      sG[dra * 64 + nca] = Cmat[(size_t)(d0 + dra) * 64 + nca] * hA[i];
      int drb = ri_b * 16 + i + hf * 8, ncb = ci_b * 16 + m;
      sG[drb * 64 + ncb] = Cmat[(size_t)(d0 + drb) * 64 + ncb] * hB[i];
    }

#if defined(USE_ASYNC_LDS)
    // wait for the *current* chunk's 4 async loads (issued last iteration /
    // prologue); the 4 just issued for c+1 stay in flight (in-order retire).
    if (c + 1 < NCHUNK) __builtin_amdgcn_s_wait_asynccnt(4);
    else                __builtin_amdgcn_s_wait_asynccnt(0);
#endif
    __syncthreads();   // sU(current) + sG now visible to all waves

    // ---- GEMM1: y_carry = W @ g^T   (M=tau, N=d, K=n) ----
    v8f ya = {}, yb = {};
#pragma unroll
    for (int kk = 0; kk < 16; ++kk) {
      int k0 = kk * 4 + hf * 2;
      v2f a, bf;
      a.x = Wfa[kk][0]; a.y = Wfa[kk][1];
      bf.x = sG[(ci_a * 16 + m) * 64 + k0];
      bf.y = sG[(ci_a * 16 + m) * 64 + k0 + 1];
      ya = wmma_f32(a, bf, ya);
      a.x = Wfb[kk][0]; a.y = Wfb[kk][1];
      bf.x = sG[(ci_b * 16 + m) * 64 + k0];
      bf.y = sG[(ci_b * 16 + m) * 64 + k0 + 1];
      yb = wmma_f32(a, bf, yb);
    }

    // ---- intra-chunk causal Toeplitz: y[tau] += sum_{j<=tau} K[tau-j]*u[j] ----
    {
      int tbase = ri_a * 16 + hf * 8;    // uniform per 16-lane half
      int dd = ci_a * 16 + m;
      for (int j = 0; j <= tbase + 7; ++j) {
        float uj = sU[j * 64 + dd];
        int imin = j - tbase; if (imin < 0) imin = 0;   // uniform per 16-lane half
        for (int i = imin; i < 8; ++i)
          ya[i] = fmaf(sK[(tbase + i - j) * 64 + dd], uj, ya[i]);
      }
    }
    {
      int tbase = ri_b * 16 + hf * 8;
      int dd = ci_b * 16 + m;
      for (int j = 0; j <= tbase + 7; ++j) {
        float uj = sU[j * 64 + dd];
        int imin = j - tbase; if (imin < 0) imin = 0;
        for (int i = imin; i < 8; ++i)
          yb[i] = fmaf(sK[(tbase + i - j) * 64 + dd], uj, yb[i]);
      }
    }

    // ---- D-skip, freq_align scale, reward@t==0; store pre-LN y ----
    float* obase = y_out + ((size_t)b * SEQ_L + (size_t)c * CHUNK) * D_MODEL + d0;
#pragma unroll
    for (int i = 0; i < 8; ++i) {
      int tau = ri_a * 16 + i + hf * 8;
      int dd = ci_a * 16 + m;
      float val = (ya[i] + sU[tau * 64 + dd] * Dva) * FAa;
      val += (c == 0 && tau == 0) ? RWa : 0.0f;
      obase[(size_t)tau * D_MODEL + dd] = val;
    }
#pragma unroll
    for (int i = 0; i < 8; ++i) {
      int tau = ri_b * 16 + i + hf * 8;
      int dd = ci_b * 16 + m;
      float val = (yb[i] + sU[tau * 64 + dd] * Dvb) * FAb;
      val += (c == 0 && tau == 0) ? RWb : 0.0f;
      obase[(size_t)tau * D_MODEL + dd] = val;
    }

    // ---- state update: h = rT.*h + u^T @ V  (M=d, N=n, K=j) ----
#pragma unroll
    for (int i = 0; i < 8; ++i) { hA[i] *= rta; hB[i] *= rtb; }
#pragma unroll
    for (int kk = 0; kk < 16; ++kk) {
      int k0 = kk * 4 + hf * 2;
      v2f a, bf;
      a.x = sU[(k0)     * 64 + ri_a * 16 + m];
      a.y = sU[(k0 + 1) * 64 + ri_a * 16 + m];
      bf.x = Vfa[kk][0]; bf.y = Vfa[kk][1];
      hA = wmma_f32(a, bf, hA);
      a.x = sU[(k0)     * 64 + ri_b * 16 + m];
      a.y = sU[(k0 + 1) * 64 + ri_b * 16 + m];
      bf.x = Vfb[kk][0]; bf.y = Vfb[kk][1];
      hB = wmma_f32(a, bf, hB);
    }
    __syncthreads();   // all reads of sU/sG done before reuse next iteration
  }
}

// ---------------------------------------------------------------------------
// Kernel 4: LayerNorm over D, in place on d_out. One block per (b,t) row.
// ---------------------------------------------------------------------------
__global__ __launch_bounds__(256) void ssm_ln(const float* __restrict__ gamma,
                                              const float* __restrict__ beta,
                                              float* __restrict__ y) {
  __shared__ float rsum[256];
  __shared__ float rsq[256];
  int tid = threadIdx.x;
  float* p = y + (size_t)blockIdx.x * D_MODEL;
  float4 x = *(float4*)(p + tid * 4);
  rsum[tid] = x.x + x.y + x.z + x.w;
  rsq[tid] = x.x * x.x + x.y * x.y + x.z * x.z + x.w * x.w;
  __syncthreads();
  for (int off = 128; off > 0; off >>= 1) {
    if (tid < off) { rsum[tid] += rsum[tid + off]; rsq[tid] += rsq[tid + off]; }
    __syncthreads();
  }
  float mean = rsum[0] * (1.0f / 1024.0f);
  float var = rsq[0] * (1.0f / 1024.0f) - mean * mean;
  float inv = rsqrtf(var + 1e-5f);
  float4 g4 = *(const float4*)(gamma + tid * 4);
  float4 b4 = *(const float4*)(beta + tid * 4);
  float4 o;
  o.x = (x.x - mean) * inv * g4.x + b4.x;
  o.y = (x.y - mean) * inv * g4.y + b4.y;
  o.z = (x.z - mean) * inv * g4.z + b4.z;
  o.w = (x.w - mean) * inv * g4.w + b4.w;
  *(float4*)(p + tid * 4) = o;
}

// ---------------------------------------------------------------------------
// Kernel 5: final_state[b,n] = coef[n] * sum_d B[n,d]*u[b,L-1,d]
// ---------------------------------------------------------------------------
__global__ __launch_bounds__(256) void ssm_final(const float* __restrict__ u,
                                                 const float* __restrict__ Bm,
                                                 const float* __restrict__ ws,
                                                 float* __restrict__ out_fs) {
  int tid = threadIdx.x;
  int b = tid >> 6, n = tid & 63;
  const float* ul = u + ((size_t)b * SEQ_L + (SEQ_L - 1)) * D_MODEL;
  const float* br = Bm + (size_t)n * D_MODEL;
  float acc = 0.0f;
  for (int d = 0; d < D_MODEL; ++d) acc = fmaf(br[d], ul[d], acc);
  out_fs[b * 64 + n] = acc * ws[WS_COEF + n];
}

// ---------------------------------------------------------------------------
extern "C" void kernel_launch(void* const* d_in, const int* in_sizes, int n_in,
                              void* d_out, int out_size, void* d_ws,
                              size_t ws_size, hipStream_t stream) {
  const float* u      = (const float*)d_in[0];
  const float* A_log  = (const float*)d_in[1];
  const float* Bm     = (const float*)d_in[2];
  const float* Cm     = (const float*)d_in[3];
  const float* Dv     = (const float*)d_in[4];
  const float* delta  = (const float*)d_in[5];
  const float* gamma  = (const float*)d_in[6];
  const float* beta   = (const float*)d_in[7];
  const float* fa     = (const float*)d_in[8];
  const float* rw     = (const float*)d_in[9];
  float* ws = (float*)d_ws;
  float* y = (float*)d_out;
  float* fs = y + (size_t)BATCH * SEQ_L * D_MODEL;

  ssm_setup<<<1, 256, 0, stream>>>(A_log, delta, ws);
  ssm_ksub<<<32, 256, 0, stream>>>(Cm, ws);
  ssm_main<<<BATCH * 16, 256, 0, stream>>>(u, Cm, Dv, fa, rw, ws, y);
  ssm_ln<<<BATCH * SEQ_L, 256, 0, stream>>>(gamma, beta, y);
  ssm_final<<<1, 256, 0, stream>>>(u, Bm, ws, fs);
}